// Tokenizer_48206712930605
// MI455X (gfx1250) — compile-verified
//
#include <hip/hip_runtime.h>
#include <cstdint>

// ---------------------------------------------------------------------------
// Tokenization-lattice forward/backward logsumexp DP for gfx1250 (MI455X).
//
// Memory-bound (~170MB traffic -> ~7us at 23.3 TB/s); compute is ~17M exp/log.
// The DP is a log-semiring scan (no matmul structure), so the main kernel uses
// the CDNA5 paths the chip rewards there: async global->LDS staging of the
// 128KB weight table (ASYNCcnt + s_wait_asynccnt), LDS random gathers, and
// wave32 ds-shuffle butterfly logsumexp. The entropy reduction ent = rowsum(H)
// IS a matrix op (H[64,32] x ones) and is done with V_WMMA_F32_16X16X4_F32.
// ---------------------------------------------------------------------------

#define NBv     64
#define NBLKv   32
#define MDIM    16
#define LDIM    256
#define VDIM    32000
#define BROWS   (NBv * NBLKv)          // 2048 lattice rows
#define EDGES   (MDIM * LDIM)          // 4096 edges per row
#define FINF    1000000000.0f

#define ROWS_PB 8                      // rows per block (one 16-lane half-wave each)
#define THREADS 128                    // 4 waves -> 8 half-waves
#define LP      260                    // padded node-array pitch (L+1 = 257)
#define SMEM_BYTES (VDIM * 4 + 2 * ROWS_PB * LP * 4)   // 144,640 B < 320 KB

#if defined(__has_builtin)
#if __has_builtin(__builtin_amdgcn_global_load_async_to_lds_b128)
#define HAVE_ASYNC_B128 1
#endif
#if __has_builtin(__builtin_amdgcn_s_wait_asynccnt)
#define HAVE_WAIT_ASYNC 1
#endif
#endif

#define AS_GLOBAL __attribute__((address_space(1)))
#define AS_LDS    __attribute__((address_space(3)))
typedef int   v4i_t __attribute__((ext_vector_type(4)));
typedef float v2f   __attribute__((ext_vector_type(2)));
typedef float v8f   __attribute__((ext_vector_type(8)));

__device__ __forceinline__ float half16_max(float v) {
    // butterfly over the 16-lane half (xor 1/2/4/8 never crosses bit 4)
    v = fmaxf(v, __shfl_xor(v, 1));
    v = fmaxf(v, __shfl_xor(v, 2));
    v = fmaxf(v, __shfl_xor(v, 4));
    v = fmaxf(v, __shfl_xor(v, 8));
    return v;
}

__global__ __launch_bounds__(THREADS)
void lattice_kernel(const int* __restrict__ ids, const float* __restrict__ ms,
                    const float* __restrict__ wts,
                    float* __restrict__ outA, float* __restrict__ outM,
                    float* __restrict__ outC, float* __restrict__ Hws) {
    extern __shared__ float smem[];
    float* s_w     = smem;                     // [VDIM] weight table
    float* s_alpha = smem + VDIM;              // [ROWS_PB][LP] alpha nodes
    float* s_beta  = s_alpha + ROWS_PB * LP;   // [ROWS_PB][LP] beta nodes

    const int tid = threadIdx.x;

    // ---- Stage weight table into LDS (async global->LDS, ASYNCcnt) ----
#if HAVE_ASYNC_B128
    for (int i = tid; i < VDIM / 4; i += THREADS) {
        __builtin_amdgcn_global_load_async_to_lds_b128(
            (AS_GLOBAL v4i_t*)(wts + 4 * i),
            (AS_LDS    v4i_t*)(s_w + 4 * i), 0, 0);
    }
#if HAVE_WAIT_ASYNC
    __builtin_amdgcn_s_wait_asynccnt(0);
#else
    asm volatile("s_wait_asynccnt 0" ::: "memory");
#endif
#else
    for (int i = tid; i < VDIM / 4; i += THREADS)
        ((float4*)s_w)[i] = ((const float4*)wts)[i];
#endif
    __syncthreads();

    const int wave = tid >> 5;
    const int half = (tid >> 4) & 1;
    const int m    = tid & 15;               // window slot owned by this lane
    const int rl   = wave * 2 + half;        // row-in-block
    const int row  = blockIdx.x * ROWS_PB + rl;
    const int*   idr = ids + (size_t)row * EDGES;
    const float* msr = ms  + (size_t)row * EDGES;

    // ---------------- Forward scan: alpha + entropy ----------------
    float awin = (m == 0) ? 0.f : -FINF;     // awin = alpha[l - m]
    float hwin = 0.f;                        // hwin = h[l - m]
    float na = 0.f, nh = 0.f;
    if (m == 0) s_alpha[rl * LP + 0] = 0.f;

    for (int l0 = 0; l0 < LDIM; l0 += 4) {
        int4   id4 = *(const int4*)  (idr + m * LDIM + l0);
        float4 ms4 = *(const float4*)(msr + m * LDIM + l0);
#pragma unroll
        for (int j = 0; j < 4; ++j) {
            int   idv = ((const int*)&id4)[j];
            float msv = ((const float*)&ms4)[j];
            float ts  = s_w[idv];
            float w   = ts * msv - FINF * (1.f - msv);
            float sc  = awin + w;
            float mx  = half16_max(sc);
            float e   = __expf(sc - mx);
            float cb  = e * (hwin - ts);
            float se = e, sq = cb;
            se += __shfl_xor(se, 1); sq += __shfl_xor(sq, 1);
            se += __shfl_xor(se, 2); sq += __shfl_xor(sq, 2);
            se += __shfl_xor(se, 4); sq += __shfl_xor(sq, 4);
            se += __shfl_xor(se, 8); sq += __shfl_xor(sq, 8);
            na = mx + __logf(se);            // alpha[l+1]
            nh = sq / se;                    // h[l+1]
            float ash = __shfl_up(awin, 1, 16);
            float hsh = __shfl_up(hwin, 1, 16);
            awin = (m == 0) ? na : ash;
            hwin = (m == 0) ? nh : hsh;
            if (m == 0) s_alpha[rl * LP + l0 + j + 1] = na;
        }
    }
    if (m == 0) Hws[row] = nh + na;          // H = h[L] + logZ

    // ---------------- Backward scan: beta ----------------
    float bwin = (m == 0) ? 0.f : -FINF;     // bwin = beta[l + 1 + m]
    if (m == 0) s_beta[rl * LP + LDIM] = 0.f;
    for (int l = LDIM - 1; l >= 0; --l) {
        int col = l + m;                     // diagonal access w[m, l+m]
        float w = -FINF;
        if (col < LDIM) {
            float msv = msr[m * LDIM + col];
            float ts  = s_w[idr[m * LDIM + col]];
            w = ts * msv - FINF * (1.f - msv);
        }
        float sc = bwin + w;
        float mx = half16_max(sc);
        float e  = __expf(sc - mx);
        float se = e;
        se += __shfl_xor(se, 1);
        se += __shfl_xor(se, 2);
        se += __shfl_xor(se, 4);
        se += __shfl_xor(se, 8);
        float nb = mx + __logf(se);          // beta[l]
        float bsh = __shfl_up(bwin, 1, 16);
        bwin = (m == 0) ? nb : bsh;
        if (m == 0) s_beta[rl * LP + l] = nb;
    }
    __syncthreads();

    // ---------------- Outputs: coalesced full-wave float4 streaming ---------
    const int lane = tid & 31;
#pragma unroll
    for (int rr = 0; rr < 2; ++rr) {
        const int orl  = wave * 2 + rr;
        const int orow = blockIdx.x * ROWS_PB + orl;
        const int*   oid = ids + (size_t)orow * EDGES;
        const float* oms = ms  + (size_t)orow * EDGES;
        const float* sa  = s_alpha + orl * LP;
        const float* sb  = s_beta  + orl * LP;
        const float  logZ = sa[LDIM];
        for (int e0 = lane * 4; e0 < EDGES; e0 += 32 * 4) {
            int4   id4 = *(const int4*)  (oid + e0);
            float4 ms4 = *(const float4*)(oms + e0);
            float4 va, vm, vc;
#pragma unroll
            for (int j = 0; j < 4; ++j) {
                int   ee  = e0 + j;
                int   mm  = ee >> 8;         // edge length index
                int   ll  = ee & 255;        // position
                int   idv = ((const int*)&id4)[j];
                float msv = ((const float*)&ms4)[j];
                float ts  = s_w[idv];
                bool  valid = msv > 0.f;
                float aS = -FINF, bS = -FINF;
                if (ll >= mm) { aS = sa[ll - mm]; bS = sb[ll - mm]; }
                float bE     = sb[ll + 1];
                float medge  = valid ? (aS + ts + bE - logZ) : -FINF;
                float nstart = (ll >= mm) ? (aS + bS - logZ) : -FINF;
                float cv     = valid ? (medge - nstart) : -FINF;
                ((float*)&va)[j] = fminf(medge, 0.f);
                ((float*)&vm)[j] = medge;
                ((float*)&vc)[j] = cv;
            }
            size_t ge = (size_t)orow * EDGES + e0;
            *(float4*)(outA + ge) = va;
            *(float4*)(outM + ge) = vm;
            *(float4*)(outC + ge) = vc;
        }
    }
}

// ent[i] = max(sum_j H[i,j], 0) as H[64,32] x ones via V_WMMA_F32_16X16X4_F32.
// One full wave, EXEC all-ones (WMMA requirement), uniform control flow.
// A 16x4 f32 layout (ISA 7.12.2): lanes 0-15 hold M=lane, K=0,1 in 2 VGPRs;
// lanes 16-31 hold M=lane-16, K=2,3. B = ones is layout-invariant.
// C/D 16x16 f32 layout: lane n (0-15) VGPR v = D[v][n]; lane 16+n VGPR v = D[8+v][n].
__global__ __launch_bounds__(32)
void ent_kernel(const float* __restrict__ H, float* __restrict__ ent) {
    const int lane = threadIdx.x;
    const int mrow = lane & 15;
    const int khi  = lane >> 4;                  // 0 -> K=0,1 ; 1 -> K=2,3
    v2f b; b.x = 1.f; b.y = 1.f;                 // all-ones B (4x16)
#pragma unroll
    for (int c = 0; c < NBv / 16; ++c) {         // 4 chunks of 16 ent rows
        v8f acc = {};
#pragma unroll
        for (int t = 0; t < NBLKv / 4; ++t) {    // 8 K-steps of 4 -> 32 cols
            const float* src = H + (size_t)(c * 16 + mrow) * NBLKv + 4 * t + 2 * khi;
            v2f a; a.x = src[0]; a.y = src[1];
            acc = __builtin_amdgcn_wmma_f32_16x16x4_f32(
                false, a, false, b, (short)0, acc, false, false);
        }
        // D[m][n] identical across n: lane 0 carries rows 0-7, lane 16 rows 8-15.
        if (mrow == 0) {
#pragma unroll
            for (int v = 0; v < 8; ++v)
                ent[c * 16 + khi * 8 + v] = fmaxf(acc[v], 0.f);
        }
    }
}

extern "C" void kernel_launch(void* const* d_in, const int* in_sizes, int n_in,
                              void* d_out, int out_size, void* d_ws, size_t ws_size,
                              hipStream_t stream) {
    const int*   ids = (const int*)d_in[0];     // fwd_ids (NB,NBLK,M,L) int32
    const float* msp = (const float*)d_in[1];   // fwd_ms  (NB,NBLK,M,L) f32
    // d_in[2] = lengths (unused by the reference)
    const float* wts = (const float*)d_in[3];   // weights (V,1) f32

    float* out    = (float*)d_out;
    float* outEnt = out;                         // [64]
    float* outA   = out + NBv;                   // [B*M*L]
    float* outM   = outA + (size_t)BROWS * EDGES;
    float* outC   = outM + (size_t)BROWS * EDGES;
    float* Hws    = (float*)d_ws;                // [BROWS] per-row entropy

    (void)hipFuncSetAttribute((const void*)lattice_kernel,
                              hipFuncAttributeMaxDynamicSharedMemorySize,
                              SMEM_BYTES);
    lattice_kernel<<<BROWS / ROWS_PB, THREADS, SMEM_BYTES, stream>>>(
        ids, msp, wts, outA, outM, outC, Hws);
    ent_kernel<<<1, 32, 0, stream>>>(Hws, outEnt);
}